// GNN_27092653703483
// MI455X (gfx1250) — compile-verified
//
#include <hip/hip_runtime.h>
#include <hip/hip_bf16.h>

#define N_NODES 50000
#define N_EDGES 800000
#define DIM     256
#define N_LAYERS 3
#define BN_EPS  1e-5f
#define LDS_STRIDE 260   // 260 % 64 == 4 -> conflict-free A-fragment reads

typedef float v2f __attribute__((ext_vector_type(2)));
typedef float v8f __attribute__((ext_vector_type(8)));

// ---------------------------------------------------------------- zero fill
__global__ __launch_bounds__(256) void zero_f4(float4* __restrict__ p, long long n4) {
    long long i = (long long)blockIdx.x * blockDim.x + threadIdx.x;
    if (i < n4) p[i] = make_float4(0.f, 0.f, 0.f, 0.f);
}

// ---------------------------------------------------------------- in-degree
__global__ __launch_bounds__(256) void deg_kernel(const long long* __restrict__ dst,
                                                  float* __restrict__ deg, int e) {
    int i = blockIdx.x * blockDim.x + threadIdx.x;
    if (i < e) atomicAdd(&deg[dst[i]], 1.0f);
}

// ------------------------------------------------- edge scatter: agg += x[src]
// 64 threads per edge, float4 gather + 4 f32 atomics (global_atomic_add_f32)
__global__ __launch_bounds__(256) void scatter_kernel(const float* __restrict__ x,
                                                      const long long* __restrict__ src,
                                                      const long long* __restrict__ dst,
                                                      float* __restrict__ agg,
                                                      long long nwork) {
    long long i = (long long)blockIdx.x * blockDim.x + threadIdx.x;
    if (i >= nwork) return;
    long long e = i >> 6;
    int j = (int)(i & 63) * 4;
    long long s = src[e];
    long long d = dst[e];
    float4 v = *(const float4*)(x + s * DIM + j);
    float* p = agg + d * DIM + j;
    atomicAdd(p + 0, v.x);
    atomicAdd(p + 1, v.y);
    atomicAdd(p + 2, v.z);
    atomicAdd(p + 3, v.w);
}

// ------------------------------------------------------------- WMMA GEMM
// h = (agg/deg) @ Wl + bl + x @ Wr      (h may alias agg: block reads only its
// own 16 rows into LDS before writing them back)
// Block = 256 thr (8 waves), covers 16 rows x 256 cols; wave w -> cols [32w,32w+32)
// A-panels staged via GLOBAL_LOAD_ASYNC_TO_LDS_B128 (ASYNCcnt DMA path).
__global__ __launch_bounds__(256) void gemm_wmma(const float* __restrict__ xsrc,
                                                 const float* __restrict__ agg,
                                                 const float* __restrict__ deg,
                                                 const float* __restrict__ Wl,
                                                 const float* __restrict__ bl,
                                                 const float* __restrict__ Wr,
                                                 float* __restrict__ h) {
    __shared__ float lds_x[16 * LDS_STRIDE];
    __shared__ float lds_a[16 * LDS_STRIDE];

    const int tid = threadIdx.x;
    const int r0  = blockIdx.x * 16;           // 50000 = 16 * 3125, no guards

    // ---- async-DMA stage 16 rows x 256 cols of x and agg into LDS ----
    // 256 lanes x 16B = 4KB per issue; 4 issues per 16KB panel.
    {
        int row = tid >> 4;                    // 16 threads per row
        int c   = (tid & 15) * 4;              // float col, +64 per issue
        int g   = r0 + row;
        const float* xr = xsrc + (size_t)g * DIM;
        const float* ar = agg  + (size_t)g * DIM;
        unsigned lx = (unsigned)(uintptr_t)&lds_x[row * LDS_STRIDE + c];
        unsigned la = (unsigned)(uintptr_t)&lds_a[row * LDS_STRIDE + c];
#pragma unroll
        for (int it = 0; it < 4; ++it) {
            asm volatile("global_load_async_to_lds_b128 %0, %1, off"
                         :: "v"(lx + it * 64 * 4), "v"(xr + c + it * 64)
                         : "memory");
            asm volatile("global_load_async_to_lds_b128 %0, %1, off"
                         :: "v"(la + it * 64 * 4), "v"(ar + c + it * 64)
                         : "memory");
        }
        asm volatile("s_wait_asynccnt 0" ::: "memory");
    }
    __syncthreads();

    const int wave = tid >> 5;
    const int lane = tid & 31;
    const int half = lane >> 4;                // 0: K=k0,k0+1  1: K=k0+2,k0+3
    const int l15  = lane & 15;
    const int col0 = wave * 32;
    const int kb   = half * 2;
    const int arow = l15 * LDS_STRIDE;

    // mean-normalization folded into the A(agg) fragment: A row M == lane&15
    const float inv = 1.0f / fmaxf(deg[r0 + l15], 1.0f);

    // C/D: VGPR r holds (M = r + 8*half, N = l15); init with bias -> fused +bl
    v8f acc[2];
#pragma unroll
    for (int t = 0; t < 2; ++t) {
        float b = bl[col0 + t * 16 + l15];
#pragma unroll
        for (int r = 0; r < 8; ++r) acc[t][r] = b;
    }

    for (int k0 = 0; k0 < DIM; k0 += 4) {
        v2f aA, aX;
        int o = arow + k0 + kb;
        aA.x = lds_a[o] * inv; aA.y = lds_a[o + 1] * inv;
        aX.x = lds_x[o];       aX.y = lds_x[o + 1];
        const float* wl = Wl + (size_t)(k0 + kb) * DIM + col0 + l15;
        const float* wr = Wr + (size_t)(k0 + kb) * DIM + col0 + l15;
#pragma unroll
        for (int t = 0; t < 2; ++t) {
            v2f bL, bR;
            bL.x = wl[t * 16]; bL.y = wl[DIM + t * 16];
            bR.x = wr[t * 16]; bR.y = wr[DIM + t * 16];
            acc[t] = __builtin_amdgcn_wmma_f32_16x16x4_f32(
                false, aA, false, bL, (short)0, acc[t], false, false);
            acc[t] = __builtin_amdgcn_wmma_f32_16x16x4_f32(
                false, aX, false, bR, (short)0, acc[t], false, false);
        }
    }

#pragma unroll
    for (int r = 0; r < 8; ++r) {
        int g = r0 + half * 8 + r;
        float* hp = h + (size_t)g * DIM + col0 + l15;
        hp[0]  = acc[0][r];
        hp[16] = acc[1][r];
    }
}

// ------------------------------------------------------------ BN column stats
__global__ __launch_bounds__(256) void bn_reduce(const float* __restrict__ h,
                                                 float* __restrict__ sums,
                                                 float* __restrict__ sumsq, int n) {
    int col = threadIdx.x;
    float s = 0.f, s2 = 0.f;
    for (int row = blockIdx.x; row < n; row += gridDim.x) {
        float v = h[(size_t)row * DIM + col];
        s += v;
        s2 += v * v;
    }
    atomicAdd(&sums[col], s);
    atomicAdd(&sumsq[col], s2);
}

__global__ void bn_finalize(const float* __restrict__ sums,
                            const float* __restrict__ sumsq,
                            const float* __restrict__ gamma,
                            const float* __restrict__ beta,
                            float* __restrict__ scale,
                            float* __restrict__ shift, float invN) {
    int t = threadIdx.x;
    float mu  = sums[t] * invN;
    float var = sumsq[t] * invN - mu * mu;   // biased variance, matches reference
    float rs  = rsqrtf(var + BN_EPS);
    float sc  = gamma[t] * rs;
    scale[t] = sc;
    shift[t] = beta[t] - mu * sc;
}

// ------------------------------------------------------- normalize + ReLU
__global__ __launch_bounds__(256) void norm_relu(const float4* __restrict__ h,
                                                 const float* __restrict__ scale,
                                                 const float* __restrict__ shift,
                                                 float4* __restrict__ out, long long n4) {
    long long i = (long long)blockIdx.x * blockDim.x + threadIdx.x;
    if (i >= n4) return;
    int col = (int)(i & 63) * 4;             // (4*i) % 256
    float4 v = h[i];
    float4 o;
    o.x = fmaxf(0.f, v.x * scale[col + 0] + shift[col + 0]);
    o.y = fmaxf(0.f, v.y * scale[col + 1] + shift[col + 1]);
    o.z = fmaxf(0.f, v.z * scale[col + 2] + shift[col + 2]);
    o.w = fmaxf(0.f, v.w * scale[col + 3] + shift[col + 3]);
    out[i] = o;
}

// ---------------------------------------------------------------- launcher
extern "C" void kernel_launch(void* const* d_in, const int* in_sizes, int n_in,
                              void* d_out, int out_size, void* d_ws, size_t ws_size,
                              hipStream_t stream) {
    const float*     x     = (const float*)d_in[0];
    const float*     Wl    = (const float*)d_in[1];
    const float*     bl    = (const float*)d_in[2];
    const float*     Wr    = (const float*)d_in[3];
    const float*     gamma = (const float*)d_in[4];
    const float*     beta  = (const float*)d_in[5];
    const long long* ei    = (const long long*)d_in[6];
    const long long* src   = ei;
    const long long* dst   = ei + N_EDGES;

    const size_t ND = (size_t)N_NODES * DIM;       // 12.8M floats
    float* ws    = (float*)d_ws;
    float* agg   = ws;                              // also serves as h (in-place)
    float* buf0  = ws + ND;
    float* buf1  = buf0 + ND;
    float* deg   = buf1 + ND;
    float* sums  = deg + N_NODES;
    float* sumsq = sums + DIM;
    float* scale = sumsq + DIM;
    float* shift = scale + DIM;

    const long long nd4   = (long long)(ND / 4);               // 3.2M float4
    const int       ndBlk = (int)((nd4 + 255) / 256);          // 12500
    const long long ew    = (long long)N_EDGES * 64;           // scatter work items
    const int       ewBlk = (int)((ew + 255) / 256);           // 200000

    // ---- in-degree (once, reused by all layers) ----
    zero_f4<<<(N_NODES / 4 + 255) / 256, 256, 0, stream>>>((float4*)deg, N_NODES / 4);
    deg_kernel<<<(N_EDGES + 255) / 256, 256, 0, stream>>>(dst, deg, N_EDGES);

    for (int l = 0; l < N_LAYERS; ++l) {
        const float* xin  = (l == 0) ? x : ((l == 1) ? buf0 : buf1);
        float*       xout = (l == 0) ? buf0 : ((l == 1) ? buf1 : (float*)d_out);

        // agg = 0 ; agg += x[src] scattered by dst
        zero_f4<<<ndBlk, 256, 0, stream>>>((float4*)agg, nd4);
        scatter_kernel<<<ewBlk, 256, 0, stream>>>(xin, src, dst, agg, ew);

        // h = (agg/deg)@Wl + bl + xin@Wr   (h aliases agg, per-block in-place)
        gemm_wmma<<<N_NODES / 16, 256, 0, stream>>>(xin, agg, deg,
                                                    Wl + (size_t)l * DIM * DIM,
                                                    bl + (size_t)l * DIM,
                                                    Wr + (size_t)l * DIM * DIM,
                                                    agg);

        // batch-norm stats + fused normalize/ReLU
        zero_f4<<<1, 128, 0, stream>>>((float4*)sums, 128);    // sums+sumsq = 512 f
        bn_reduce<<<256, 256, 0, stream>>>(agg, sums, sumsq, N_NODES);
        bn_finalize<<<1, DIM, 0, stream>>>(sums, sumsq,
                                           gamma + (size_t)l * DIM,
                                           beta + (size_t)l * DIM,
                                           scale, shift, 1.0f / (float)N_NODES);
        norm_relu<<<ndBlk, 256, 0, stream>>>((const float4*)agg, scale, shift,
                                             (float4*)xout, nd4);
    }
}